// GraphEncoderWithResidual_10857677324491
// MI455X (gfx1250) — compile-verified
//
#include <hip/hip_runtime.h>

// GraphSAGE (2-layer, mean aggr) + residual, for MI455X (gfx1250, wave32).
//
// Pipeline:
//  1) zero deg+agg                (memset-style kernel)
//  2) deg[dst] += 1               (edge-parallel atomics)
//  3) deg <- 1/max(deg,1)
//  4) agg[dst] += x[src]          (wave-per-edge, float4 atomics)
//  5) h1 = relu(x@W1l^T + b1 + (agg*invdeg)@W1r^T)          [WMMA f32 16x16x4]
//  6) zero agg
//  7) agg[dst] += h1[src]
//  8) out = relu([h1|agg*invdeg]@W2^T + b2)@Wlin^T + blin
//           + x@Wsc^T + bsc        (fused via LDS h2 tile)   [WMMA f32 16x16x4]

typedef float v2f __attribute__((ext_vector_type(2)));
typedef float v8f __attribute__((ext_vector_type(8)));

#define WMMA_F32(a, b, c) \
  __builtin_amdgcn_wmma_f32_16x16x4_f32(false, (a), false, (b), (short)0, (c), false, false)

// ---------------------------------------------------------------- utilities

__global__ __launch_bounds__(256) void k_zero(float* __restrict__ p, size_t n) {
  size_t i = (size_t)blockIdx.x * 256 + threadIdx.x;
  if (i < n) p[i] = 0.0f;
}

__global__ __launch_bounds__(256) void k_deg(const long long* __restrict__ dst,
                                             float* __restrict__ deg, int E) {
  int e = blockIdx.x * 256 + threadIdx.x;
  if (e < E) atomicAdd(&deg[(size_t)dst[e]], 1.0f);
}

__global__ __launch_bounds__(256) void k_invdeg(float* __restrict__ deg, int N) {
  int i = blockIdx.x * 256 + threadIdx.x;
  if (i < N) deg[i] = 1.0f / fmaxf(deg[i], 1.0f);
}

// Wave-per-edge scatter-add: 32 lanes x float4 = 128 floats per edge.
__global__ __launch_bounds__(256) void k_scatter(const float* __restrict__ feat,
                                                 const long long* __restrict__ src,
                                                 const long long* __restrict__ dst,
                                                 float* __restrict__ agg, int E) {
  const int C = 128;
  int e = blockIdx.x * 8 + (threadIdx.x >> 5);
  if (e >= E) return;
  int lane = threadIdx.x & 31;
  long long s = src[e];
  long long d = dst[e];
  const float4* fp = (const float4*)(feat + (size_t)s * C);
  float4 v = fp[lane];
  float* ap = agg + (size_t)d * C + lane * 4;
  atomicAdd(ap + 0, v.x);
  atomicAdd(ap + 1, v.y);
  atomicAdd(ap + 2, v.z);
  atomicAdd(ap + 3, v.w);
}

// ------------------------------------------------ layer 1: h1 = relu(SAGE(x))
// Block = 16 rows x 128 cols; 8 waves; wave w computes the 16x16 tile at col 16w.

__global__ __launch_bounds__(256) void k_sage1(const float* __restrict__ x,
                                               const float* __restrict__ agg,
                                               const float* __restrict__ invdeg,
                                               const float* __restrict__ Wl,
                                               const float* __restrict__ bl,
                                               const float* __restrict__ Wr,
                                               float* __restrict__ h1) {
  const int K = 128, OC = 128;
  int wave = threadIdx.x >> 5;
  int lane = threadIdx.x & 31;
  int half = lane >> 4;   // K-pair selector within fragment
  int lr = lane & 15;     // A row / B col within tile
  int row0 = blockIdx.x * 16;
  int c0 = wave * 16;

  const float* A1 = x + (size_t)row0 * K;
  const float* A2 = agg + (size_t)row0 * K;
  const float* B1 = Wl + (size_t)c0 * K;   // B[k][n] = W[n][k]
  const float* B2 = Wr + (size_t)c0 * K;
  float s = invdeg[row0 + lr];

  v8f acc = {};
  for (int k = 0; k < K; k += 4) {
    int kk = k + 2 * half;
    v2f a, b;
    a.x = A1[lr * K + kk];  a.y = A1[lr * K + kk + 1];
    b.x = B1[lr * K + kk];  b.y = B1[lr * K + kk + 1];
    acc = WMMA_F32(a, b, acc);
  }
  for (int k = 0; k < K; k += 4) {
    int kk = k + 2 * half;
    v2f a, b;
    a.x = A2[lr * K + kk] * s;  a.y = A2[lr * K + kk + 1] * s;
    b.x = B2[lr * K + kk];      b.y = B2[lr * K + kk + 1];
    acc = WMMA_F32(a, b, acc);
  }

  float bias = bl[c0 + lr];
#pragma unroll
  for (int i = 0; i < 8; i++) {
    float v = acc[i] + bias;
    h1[(size_t)(row0 + i + 8 * half) * OC + c0 + lr] = fmaxf(v, 0.0f);
  }
}

// -------------------- layer 2 + final linear + shortcut, fused through LDS
// Stage 1: each wave computes two 16x16 h2 tiles (cols 16w and 16w+128),
//          relu, stores into padded LDS tile [16 x 256].
// Stage 2: each wave computes one 16x16 out tile: h2s@Wlin^T + x@Wsc^T + biases.

#define LDS_STRIDE 260  // 260 % 64 == 4 -> conflict-free column walks

__global__ __launch_bounds__(256) void k_sage2_out(
    const float* __restrict__ h1, const float* __restrict__ agg,
    const float* __restrict__ invdeg, const float* __restrict__ x,
    const float* __restrict__ W2l, const float* __restrict__ b2,
    const float* __restrict__ W2r, const float* __restrict__ Wlin,
    const float* __restrict__ blin, const float* __restrict__ Wsc,
    const float* __restrict__ bsc, float* __restrict__ out) {
  const int K1 = 128;  // h1 / agg / x feature dim
  const int H2 = 256;  // hidden2 dim
  const int OC = 128;  // output dim
  __shared__ float h2s[16 * LDS_STRIDE];

  int wave = threadIdx.x >> 5;
  int lane = threadIdx.x & 31;
  int half = lane >> 4;
  int lr = lane & 15;
  int row0 = blockIdx.x * 16;

  const float* A1 = h1 + (size_t)row0 * K1;
  const float* A2 = agg + (size_t)row0 * K1;
  float s = invdeg[row0 + lr];

  // ---- stage 1: h2 tile
  int c0a = 16 * wave;
  int c0b = 16 * wave + 128;
  const float* Bl_a = W2l + (size_t)c0a * K1;
  const float* Bl_b = W2l + (size_t)c0b * K1;
  const float* Br_a = W2r + (size_t)c0a * K1;
  const float* Br_b = W2r + (size_t)c0b * K1;

  v8f acc0 = {}, acc1 = {};
  for (int k = 0; k < K1; k += 4) {
    int kk = k + 2 * half;
    v2f a, b0, b1;
    a.x = A1[lr * K1 + kk];     a.y = A1[lr * K1 + kk + 1];
    b0.x = Bl_a[lr * K1 + kk];  b0.y = Bl_a[lr * K1 + kk + 1];
    b1.x = Bl_b[lr * K1 + kk];  b1.y = Bl_b[lr * K1 + kk + 1];
    acc0 = WMMA_F32(a, b0, acc0);
    acc1 = WMMA_F32(a, b1, acc1);
  }
  for (int k = 0; k < K1; k += 4) {
    int kk = k + 2 * half;
    v2f a, b0, b1;
    a.x = A2[lr * K1 + kk] * s; a.y = A2[lr * K1 + kk + 1] * s;
    b0.x = Br_a[lr * K1 + kk];  b0.y = Br_a[lr * K1 + kk + 1];
    b1.x = Br_b[lr * K1 + kk];  b1.y = Br_b[lr * K1 + kk + 1];
    acc0 = WMMA_F32(a, b0, acc0);
    acc1 = WMMA_F32(a, b1, acc1);
  }

  float biasA = b2[c0a + lr];
  float biasB = b2[c0b + lr];
#pragma unroll
  for (int i = 0; i < 8; i++) {
    int r = i + 8 * half;
    h2s[r * LDS_STRIDE + c0a + lr] = fmaxf(acc0[i] + biasA, 0.0f);
    h2s[r * LDS_STRIDE + c0b + lr] = fmaxf(acc1[i] + biasB, 0.0f);
  }
  __syncthreads();

  // ---- stage 2: out tile at col 16*wave
  int c0 = 16 * wave;
  const float* BL = Wlin + (size_t)c0 * H2;
  const float* BS = Wsc + (size_t)c0 * K1;
  const float* Ax = x + (size_t)row0 * K1;

  v8f acc = {};
  for (int k = 0; k < H2; k += 4) {
    int kk = k + 2 * half;
    v2f a, b;
    a.x = h2s[lr * LDS_STRIDE + kk];  a.y = h2s[lr * LDS_STRIDE + kk + 1];
    b.x = BL[lr * H2 + kk];           b.y = BL[lr * H2 + kk + 1];
    acc = WMMA_F32(a, b, acc);
  }
  for (int k = 0; k < K1; k += 4) {
    int kk = k + 2 * half;
    v2f a, b;
    a.x = Ax[lr * K1 + kk];  a.y = Ax[lr * K1 + kk + 1];
    b.x = BS[lr * K1 + kk];  b.y = BS[lr * K1 + kk + 1];
    acc = WMMA_F32(a, b, acc);
  }

  float bias = blin[c0 + lr] + bsc[c0 + lr];
#pragma unroll
  for (int i = 0; i < 8; i++) {
    out[(size_t)(row0 + i + 8 * half) * OC + c0 + lr] = acc[i] + bias;
  }
}

// ---------------------------------------------------------------- launcher

extern "C" void kernel_launch(void* const* d_in, const int* in_sizes, int n_in,
                              void* d_out, int out_size, void* d_ws, size_t ws_size,
                              hipStream_t stream) {
  (void)n_in; (void)out_size; (void)ws_size;

  const float* x = (const float*)d_in[0];
  const long long* ei = (const long long*)d_in[1];
  const float* W1l = (const float*)d_in[2];
  const float* b1 = (const float*)d_in[3];
  const float* W1r = (const float*)d_in[4];
  const float* W2l = (const float*)d_in[5];
  const float* b2 = (const float*)d_in[6];
  const float* W2r = (const float*)d_in[7];
  const float* Wlin = (const float*)d_in[8];
  const float* blin = (const float*)d_in[9];
  const float* Wsc = (const float*)d_in[10];
  const float* bsc = (const float*)d_in[11];
  float* out = (float*)d_out;

  const int C = 128;
  const int N = in_sizes[0] / C;     // 50000
  const int E = in_sizes[1] / 2;     // 800000
  const long long* src = ei;
  const long long* dst = ei + E;

  // workspace layout: [deg: N][agg: N*128][h1: N*128]  (16B-aligned sections)
  float* deg = (float*)d_ws;
  float* agg = deg + N;
  float* h1 = agg + (size_t)N * C;

  const int T = 256;
  size_t n_deg_agg = (size_t)N * (C + 1);
  size_t n_agg = (size_t)N * C;

  // 1) zero deg+agg (contiguous)
  k_zero<<<dim3((unsigned)((n_deg_agg + T - 1) / T)), T, 0, stream>>>(deg, n_deg_agg);
  // 2) degrees, 3) inverse
  k_deg<<<dim3((E + T - 1) / T), T, 0, stream>>>(dst, deg, E);
  k_invdeg<<<dim3((N + T - 1) / T), T, 0, stream>>>(deg, N);
  // 4) agg = scatter_add(x)
  k_scatter<<<dim3((E + 7) / 8), T, 0, stream>>>(x, src, dst, agg, E);
  // 5) h1
  k_sage1<<<dim3(N / 16), T, 0, stream>>>(x, agg, deg, W1l, b1, W1r, h1);
  // 6) re-zero agg, 7) agg = scatter_add(h1)
  k_zero<<<dim3((unsigned)((n_agg + T - 1) / T)), T, 0, stream>>>(agg, n_agg);
  k_scatter<<<dim3((E + 7) / 8), T, 0, stream>>>(h1, src, dst, agg, E);
  // 8) fused layer2 + lin + shortcut
  k_sage2_out<<<dim3(N / 16), T, 0, stream>>>(h1, agg, deg, x, W2l, b2, W2r,
                                              Wlin, blin, Wsc, bsc, out);
}